// PCEN_11759620456826
// MI455X (gfx1250) — compile-verified
//
#include <hip/hip_runtime.h>
#include <stdint.h>

typedef __attribute__((ext_vector_type(2))) float        v2f;
typedef __attribute__((ext_vector_type(8))) float        v8f;
typedef __attribute__((ext_vector_type(4))) unsigned int u32x4;
typedef __attribute__((ext_vector_type(8))) int          i32x8;
typedef __attribute__((ext_vector_type(4))) int          i32x4;

namespace {
constexpr int   kB = 64, kT = 4096, kC = 128;
constexpr float kS     = 0.025f;   // SMOOTH_COEF
constexpr float kOm    = 0.975f;   // 1 - SMOOTH_COEF
constexpr float kFloor = 1e-6f;
constexpr int   kCG    = 16;          // channels per group (WMMA N)
constexpr int   kNCG   = kC / kCG;    // 8
constexpr int   kSeg   = 4;           // segments along T
constexpr int   kSegLen = kT / kSeg;  // 1024
constexpr int   kHalo  = 1024;        // (1-s)^1024 ~ 5e-12 << fp32 eps
}

// Hardware transcendentals (base-2): v_exp_f32 / v_log_f32 / v_rcp_f32
#define EXP2F(x) __builtin_amdgcn_exp2f(x)
#define LOG2F(x) __builtin_amdgcn_logf(x)
#define RCPF(x)  __builtin_amdgcn_rcpf(x)

// Issue a TDM load of a 16(time) x 16(chan) fp32 tile (row-contiguous channels)
// into LDS at byte offset lds_byte_off. D# packed per CDNA5 ISA ch.8.
__device__ __forceinline__ void tdm_load_tile16x16_f32(unsigned int lds_byte_off,
                                                       const float* gtile) {
  unsigned long long ga = (unsigned long long)(uintptr_t)gtile;
  u32x4 g0;
  g0[0] = 1u;                                               // count=1, user mode
  g0[1] = lds_byte_off;                                     // lds_addr
  g0[2] = (unsigned int)ga;                                 // global_addr[31:0]
  g0[3] = ((unsigned int)(ga >> 32) & 0x01FFFFFFu) | (2u << 30); // addr[56:32] | type=2
  i32x8 g1;
  g1[0] = (2 << 16);                    // data_size=2 (4 bytes); no mask/pad/iterate
  g1[1] = (kC & 0xFFFF) << 16;          // tensor_dim0[15:0]=128 | barrier_addr=0
  g1[2] = ((kT & 0xFFFF) << 16) | ((kC >> 16) & 0xFFFF);  // dim1 lo | dim0 hi
  g1[3] = ((kT >> 16) & 0xFFFF) | (16 << 16);             // tile_dim0=16 | dim1 hi
  g1[4] = 16;                           // tile_dim1=16, tile_dim2=0 (2D)
  g1[5] = kC;                           // tensor_dim0_stride[31:0]=128 elements
  g1[6] = 0;                            // stride0 hi | stride1 lo
  g1[7] = 0;                            // stride1 hi
  i32x4 z4 = {0, 0, 0, 0};
  i32x8 z8 = {0, 0, 0, 0, 0, 0, 0, 0};
  __builtin_amdgcn_tensor_load_to_lds(g0, g1, z4, z4, z8, 0);
}

__global__ __launch_bounds__(32) void pcen_wmma_kernel(
    const float* __restrict__ x, const float* __restrict__ alpha,
    const float* __restrict__ delta, const float* __restrict__ root,
    float* __restrict__ out) {
  __shared__ float tiles[2][kCG * 16];   // two 1KB tiles, double buffered

  const int lane = threadIdx.x;          // wave32
  const int half = lane >> 4;            // 0 or 1 (lane half)
  const int nl   = lane & 15;            // M (A-matrix) / N (B,C,D) index

  const int bid = blockIdx.x;
  const int seg = bid % kSeg;
  const int cg  = (bid / kSeg) % kNCG;
  const int b   = bid / (kSeg * kNCG);
  const int c0  = cg * kCG;
  const int seg_start = seg * kSegLen;
  const int seg_end   = seg_start + kSegLen;

  // Per-channel params (lane and lane+16 read the same channel -> identical)
  const int   ch  = c0 + nl;
  const float al  = fminf(alpha[ch], 1.0f);
  const float dlt = delta[ch];
  const float oor = 1.0f / fmaxf(root[ch], 1.0f);
  const float l2om = LOG2F(kOm);
  const float t3  = EXP2F(oor * LOG2F(dlt));   // delta^(1/root), constant

  // A operand = lower-triangular decay matrix L[m][k] = s*(1-s)^(m-k), split
  // into four 16x4 K-slices.  A layout: lane&15 = M, lane>>4 selects K pair.
  v2f A[4];
#pragma unroll
  for (int si = 0; si < 4; ++si) {
#pragma unroll
    for (int j = 0; j < 2; ++j) {
      const int k = 4 * si + 2 * half + j;
      const int d = nl - k;
      A[si][j] = (d >= 0) ? kS * EXP2F(l2om * (float)d) : 0.0f;
    }
  }
  // Carry decay per output row: (1-s)^(M+1), C/D layout: M = v + 8*half
  float dec[8];
#pragma unroll
  for (int v = 0; v < 8; ++v)
    dec[v] = EXP2F(l2om * (float)(v + 8 * half + 1));

  const size_t strip = ((size_t)b * kT) * kC + c0;

  float carry;
  int   t_h;
  if (seg == 0) {                    // m_0 = x_0  <=>  carry m_{-1} = x_0
    t_h = 0;
    carry = x[strip + nl];
  } else {                           // halo warm-up, residual ~(1-s)^1024
    t_h = seg_start - kHalo;
    carry = 0.0f;
  }

  const unsigned int lds0 = (unsigned int)(uintptr_t)(&tiles[0][0]);
  const unsigned int lds1 = (unsigned int)(uintptr_t)(&tiles[1][0]);

  tdm_load_tile16x16_f32(lds0, x + strip + (size_t)t_h * kC);

  for (int t0 = t_h; t0 < seg_end; t0 += 16) {
    const int bi = ((t0 - t_h) >> 4) & 1;
    if (t0 + 16 < seg_end) {         // prefetch next tile into other buffer
      tdm_load_tile16x16_f32(bi ? lds0 : lds1,
                             x + strip + (size_t)(t0 + 16) * kC);
      __builtin_amdgcn_s_wait_tensorcnt(1);   // oldest (current) tile landed
    } else {
      __builtin_amdgcn_s_wait_tensorcnt(0);
    }
    const float* tile = &tiles[bi][0];

    // Accumulator seeded with carry contribution: C[M][n] = (1-s)^(M+1)*carry
    v8f acc;
#pragma unroll
    for (int v = 0; v < 8; ++v) acc[v] = dec[v] * carry;

    // Four chained K=4 fp32 WMMAs: m-tile = L(16x16) * X(16x16) + C
#pragma unroll
    for (int si = 0; si < 4; ++si) {
      v2f Bv;   // B layout: VGPR j holds row K = 4*si + 2*half + j, col = nl
      Bv[0] = tile[(4 * si + 2 * half + 0) * 16 + nl];
      Bv[1] = tile[(4 * si + 2 * half + 1) * 16 + nl];
      acc = __builtin_amdgcn_wmma_f32_16x16x4_f32(false, A[si], false, Bv,
                                                  (short)0, acc, false, false);
    }
    // Next chunk's carry = m at row M=15 (VGPR7 of lanes 16..31), broadcast
    carry = __shfl(acc[7], 16 + nl, 32);

    if (t0 >= seg_start) {           // epilogue only on owned segment
#pragma unroll
      for (int v = 0; v < 8; ++v) {
        const int   M  = v + 8 * half;
        const float xv = tile[M * 16 + nl];
        const float m  = acc[v];
        const float t1 = EXP2F(al * LOG2F(kFloor + m));       // (eps+m)^alpha
        const float t2 = EXP2F(oor * LOG2F(xv * RCPF(t1) + dlt)); // (...)^(1/root)
        out[strip + (size_t)(t0 + M) * kC + nl] = t2 - t3;
      }
    }
  }
}

extern "C" void kernel_launch(void* const* d_in, const int* in_sizes, int n_in,
                              void* d_out, int out_size, void* d_ws, size_t ws_size,
                              hipStream_t stream) {
  const float* x     = (const float*)d_in[0];
  const float* alpha = (const float*)d_in[1];
  const float* delta = (const float*)d_in[2];
  const float* root  = (const float*)d_in[3];
  float* out = (float*)d_out;

  const int blocks = kB * kNCG * kSeg;   // 64*8*4 = 2048 waves
  pcen_wmma_kernel<<<dim3(blocks), dim3(32), 0, stream>>>(x, alpha, delta, root, out);
}